// SlotAttention_66632122630723
// MI455X (gfx1250) — compile-verified
//
#include <hip/hip_runtime.h>
#include <math.h>

#define NTOK 131072
#define DIM  512
#define NSLOT 16
#define NUM_ITER 3

typedef float v2f __attribute__((ext_vector_type(2)));
typedef float v8f __attribute__((ext_vector_type(8)));

__device__ __forceinline__ v8f wmma4(v2f a, v2f b, v8f c) {
  // D = A(16x4 f32) * B(4x16 f32) + C, full fp32
  return __builtin_amdgcn_wmma_f32_16x16x4_f32(false, a, false, b, (short)0, c, false, false);
}
__device__ __forceinline__ v8f zero8() {
  v8f z = {0.f,0.f,0.f,0.f,0.f,0.f,0.f,0.f};
  return z;
}

// ---------------- slots init: slots = mu + sigma * noise ----------------
__global__ void init_slots_k(const float* __restrict__ noise, const float* __restrict__ mu,
                             const float* __restrict__ sigma, float* __restrict__ slots) {
  int i = blockIdx.x*256 + threadIdx.x;
  if (i < NSLOT*DIM) { int d = i & (DIM-1); slots[i] = mu[d] + sigma[d]*noise[i]; }
}

// ------------- fused LayerNorm + K/V projection (WMMA f32) --------------
// block = 256 thr (8 waves) handles 16 rows; waves 0-3 -> Wk cols, 4-7 -> Wv cols
__global__ __launch_bounds__(256) void kvproj_k(const float* __restrict__ emb,
    const float* __restrict__ g, const float* __restrict__ b,
    const float* __restrict__ Wk, const float* __restrict__ Wv,
    float* __restrict__ kproj, float* __restrict__ vproj) {
  __shared__ float xs[16*516];                 // padded stride 516 -> conflict-free A reads
  const int tid = threadIdx.x;
  const size_t n0 = (size_t)blockIdx.x * 16;
  for (int i = tid; i < 16*DIM; i += 256) {
    int r = i >> 9, c = i & (DIM-1);
    xs[r*516 + c] = emb[(n0 + r)*DIM + c];
  }
  __syncthreads();
  { // LayerNorm: 16 lanes per row (xor-reduce stays inside half-wave)
    int grp = tid >> 4, j = tid & 15;
    float s = 0.f, ss = 0.f;
    for (int c = j; c < DIM; c += 16) { float x = xs[grp*516 + c]; s += x; ss += x*x; }
    s += __shfl_xor(s, 8, 32); ss += __shfl_xor(ss, 8, 32);
    s += __shfl_xor(s, 4, 32); ss += __shfl_xor(ss, 4, 32);
    s += __shfl_xor(s, 2, 32); ss += __shfl_xor(ss, 2, 32);
    s += __shfl_xor(s, 1, 32); ss += __shfl_xor(ss, 1, 32);
    float mean = s * (1.f/DIM);
    float var  = ss * (1.f/DIM) - mean*mean;
    float rstd = rsqrtf(var + 1e-5f);
    for (int c = j; c < DIM; c += 16) {
      float x = xs[grp*516 + c];
      xs[grp*516 + c] = (x - mean)*rstd*g[c] + b[c];
    }
  }
  __syncthreads();
  const int w = tid >> 5, l = tid & 31, hh = l >> 4, ln16 = l & 15;
  const float* __restrict__ W = (w < 4) ? Wk : Wv;
  float* __restrict__ outp = (w < 4) ? kproj : vproj;
  const int colbase = (w & 3) * 128;
  v8f acc[8];
  #pragma unroll
  for (int t = 0; t < 8; ++t) acc[t] = zero8();
  for (int k = 0; k < DIM; k += 4) {
    v2f a = *(const v2f*)&xs[ln16*516 + k + hh*2];         // A: lane row = ln16
    const int kr = k + hh*2;
    #pragma unroll
    for (int t = 0; t < 8; ++t) {
      int col = colbase + t*16 + ln16;
      v2f bb;                                              // B: W[k][col], coalesced/half
      bb.x = W[(size_t)kr*DIM + col];
      bb.y = W[(size_t)(kr+1)*DIM + col];
      acc[t] = wmma4(a, bb, acc[t]);
    }
  }
  #pragma unroll
  for (int t = 0; t < 8; ++t) {
    int col = colbase + t*16 + ln16;
    #pragma unroll
    for (int jj = 0; jj < 8; ++jj)
      outp[(n0 + jj + hh*8)*DIM + col] = acc[t][jj];
  }
}

// ---------------- row LayerNorm for 16 slot rows ----------------
__global__ __launch_bounds__(256) void ln_rows_k(const float* __restrict__ in,
    const float* __restrict__ g, const float* __restrict__ b, float* __restrict__ out) {
  const int r = blockIdx.x, tid = threadIdx.x;
  __shared__ float as[8], bs[8];
  float x0 = in[r*DIM + tid], x1 = in[r*DIM + 256 + tid];
  float s = x0 + x1, ss = x0*x0 + x1*x1;
  for (int off = 16; off >= 1; off >>= 1) { s += __shfl_xor(s, off, 32); ss += __shfl_xor(ss, off, 32); }
  if ((tid & 31) == 0) { as[tid>>5] = s; bs[tid>>5] = ss; }
  __syncthreads();
  if (tid == 0) {
    float S = 0.f, SS = 0.f;
    for (int i = 0; i < 8; ++i) { S += as[i]; SS += bs[i]; }
    float mean = S*(1.f/DIM);
    float var  = SS*(1.f/DIM) - mean*mean;
    as[0] = mean; bs[0] = rsqrtf(var + 1e-5f);
  }
  __syncthreads();
  float mean = as[0], rstd = bs[0];
  out[r*DIM + tid]       = (x0-mean)*rstd*g[tid] + b[tid];
  out[r*DIM + 256 + tid] = (x1-mean)*rstd*g[256+tid] + b[256+tid];
}

// -------- small [16x512] @ [512x512] scalar GEMM (q-proj, slot_keys) --------
__global__ void mm512_k(const float* __restrict__ A, const float* __restrict__ W,
                        float* __restrict__ out) {
  int id = blockIdx.x*256 + threadIdx.x;        // 8192 threads
  int k = id >> 9, c = id & (DIM-1);
  const float* a = A + k*DIM;
  float s = 0.f;
  for (int d = 0; d < DIM; ++d) s += a[d]*W[(size_t)d*DIM + c];   // coalesced over c
  out[id] = s;
}

// ---- attention: logits (WMMA) + softmax over 16 slots + block row-sums ----
// each wave: one 16(n) x 16(slot) tile; D tile: M=slot row (VGPR), N=n (lane)
__global__ __launch_bounds__(256) void attn_k(const float* __restrict__ kproj,
    const float* __restrict__ q, float* __restrict__ attn, float* __restrict__ rspart) {
  __shared__ float sm[8][16];
  const int tid = threadIdx.x;
  const int w = tid >> 5, l = tid & 31, hh = l >> 4, ln16 = l & 15;
  const size_t n0 = (size_t)blockIdx.x * 128 + w*16;
  const float* __restrict__ arow = q + ln16*DIM;             // A: M=slot rows of q
  const float* __restrict__ brow = kproj + (n0 + ln16)*DIM;  // B: N=n cols = kproj rows
  v8f acc = zero8();
  for (int k = 0; k < DIM; k += 4) {
    v2f a  = *(const v2f*)&arow[k + hh*2];
    v2f bb = *(const v2f*)&brow[k + hh*2];
    acc = wmma4(a, bb, acc);
  }
  const float scale = 0.04419417382415922f;   // 512^-0.5
  float v[8];
  float mx = -1e30f;
  #pragma unroll
  for (int j = 0; j < 8; ++j) { v[j] = acc[j]*scale; mx = fmaxf(mx, v[j]); }
  mx = fmaxf(mx, __shfl_xor(mx, 16, 32));     // other 8 slots live in lane^16
  float se = 0.f;
  #pragma unroll
  for (int j = 0; j < 8; ++j) { v[j] = expf(v[j]-mx); se += v[j]; }
  se += __shfl_xor(se, 16, 32);
  float inv = 1.f/se;
  #pragma unroll
  for (int j = 0; j < 8; ++j) {
    v[j] *= inv;
    attn[(size_t)(j + hh*8)*NTOK + n0 + ln16] = v[j];        // coalesced per half-wave
  }
  #pragma unroll
  for (int j = 0; j < 8; ++j) {               // tile sum over 16 n (within half-wave)
    float t = v[j];
    t += __shfl_xor(t, 8, 32);
    t += __shfl_xor(t, 4, 32);
    t += __shfl_xor(t, 2, 32);
    t += __shfl_xor(t, 1, 32);
    if (ln16 == 0) sm[w][hh*8 + j] = t;
  }
  __syncthreads();
  if (tid < 16) {                             // deterministic per-block partial
    float s = 0.f;
    #pragma unroll
    for (int w2 = 0; w2 < 8; ++w2) s += sm[w2][tid];
    rspart[blockIdx.x*16 + tid] = s;
  }
}

__global__ void reduce_rowsum_k(const float* __restrict__ rspart, float* __restrict__ rowsum) {
  int k = blockIdx.x, tid = threadIdx.x;
  __shared__ float sh[8];
  float s = 0.f;
  for (int i = tid; i < 1024; i += 256) s += rspart[i*16 + k];
  for (int off = 16; off >= 1; off >>= 1) s += __shfl_xor(s, off, 32);
  if ((tid&31) == 0) sh[tid>>5] = s;
  __syncthreads();
  if (tid == 0) { float t = 0.f; for (int i = 0; i < 8; ++i) t += sh[i]; rowsum[k] = t; }
}

// -------- updates = attn @ v_proj, WMMA over N chunks, per-block partials --------
__global__ __launch_bounds__(256) void updates_k(const float* __restrict__ attn,
    const float* __restrict__ vproj, float* __restrict__ updpart) {
  const int tid = threadIdx.x;
  const int w = tid >> 5, l = tid & 31, hh = l >> 4, ln16 = l & 15;
  const size_t nb = (size_t)blockIdx.x * 1024;
  const int colb = w*64;
  const float* __restrict__ arow = attn + (size_t)ln16*NTOK + nb;   // A: M=slot rows
  v8f acc[4];
  #pragma unroll
  for (int t = 0; t < 4; ++t) acc[t] = zero8();
  for (int k = 0; k < 1024; k += 4) {
    v2f a = *(const v2f*)&arow[k + hh*2];
    const size_t r = (nb + k + hh*2)*DIM;
    #pragma unroll
    for (int t = 0; t < 4; ++t) {
      int col = colb + t*16 + ln16;
      v2f bb;                                  // B: v_proj rows, coalesced per half
      bb.x = vproj[r + col];
      bb.y = vproj[r + DIM + col];
      acc[t] = wmma4(a, bb, acc[t]);
    }
  }
  float* __restrict__ dst = updpart + (size_t)blockIdx.x * (NSLOT*DIM);
  #pragma unroll
  for (int t = 0; t < 4; ++t) {
    int col = colb + t*16 + ln16;
    #pragma unroll
    for (int jj = 0; jj < 8; ++jj)
      dst[(jj + hh*8)*DIM + col] = acc[t][jj];
  }
}

__global__ void reduce_upd_k(const float* __restrict__ updpart, float* __restrict__ upd) {
  int i = blockIdx.x*256 + threadIdx.x;        // < 8192
  float s = 0.f;
  for (int b = 0; b < 128; ++b) s += updpart[(size_t)b*(NSLOT*DIM) + i];
  upd[i] = s;
}

// -------- GRU gate GEMMs: gi = (upd/rowsum) @ Wih^T + bih ; gh = slots @ Whh^T + bhh --------
__global__ void gigh_k(const float* __restrict__ upd, const float* __restrict__ rowsum,
    const float* __restrict__ slots,
    const float* __restrict__ Wih, const float* __restrict__ Whh,
    const float* __restrict__ bih, const float* __restrict__ bhh,
    float* __restrict__ gi, float* __restrict__ gh) {
  int id = blockIdx.x*256 + threadIdx.x;       // < 49152
  int half = id / 24576;
  int r = id - half*24576;
  int k = r / 1536, c = r - k*1536;
  const float* a; const float* wr; float bias, sc; float* dst;
  if (half == 0) { a = upd   + k*DIM; wr = Wih + (size_t)c*DIM; bias = bih[c]; sc = 1.f/(rowsum[k] + 1e-8f); dst = gi; }
  else           { a = slots + k*DIM; wr = Whh + (size_t)c*DIM; bias = bhh[c]; sc = 1.f;                     dst = gh; }
  float s = 0.f;
  for (int d = 0; d < DIM; ++d) s += a[d]*wr[d];
  dst[k*1536 + c] = s*sc + bias;
}

__global__ void gru_elem_k(const float* __restrict__ gi, const float* __restrict__ gh,
                           float* __restrict__ slots) {
  int id = blockIdx.x*256 + threadIdx.x;       // < 8192
  int k = id >> 9, d = id & (DIM-1);
  int base = k*1536;
  float ir = gi[base + d], iz = gi[base + DIM + d], in_ = gi[base + 2*DIM + d];
  float hr = gh[base + d], hz = gh[base + DIM + d], hn  = gh[base + 2*DIM + d];
  float rr = 1.f/(1.f + expf(-(ir + hr)));
  float zz = 1.f/(1.f + expf(-(iz + hz)));
  float nn = tanhf(in_ + rr*hn);
  slots[id] = (1.f - zz)*nn + zz*slots[id];    // in-place safe (elementwise)
}

__global__ void ffn1_k(const float* __restrict__ h, const float* __restrict__ W1,
                       const float* __restrict__ b1, float* __restrict__ hid) {
  int id = blockIdx.x*256 + threadIdx.x;       // < 16384
  int k = id >> 10, c = id & 1023;
  const float* a = h + k*DIM;
  float s = 0.f;
  for (int d = 0; d < DIM; ++d) s += a[d]*W1[(size_t)d*1024 + c];
  s += b1[c];
  hid[id] = s > 0.f ? s : 0.f;
}

__global__ void ffn2_k(const float* __restrict__ hid, const float* __restrict__ W2,
                       const float* __restrict__ b2, float* __restrict__ slots) {
  int id = blockIdx.x*256 + threadIdx.x;       // < 8192
  int k = id >> 9, d = id & (DIM-1);
  const float* a = hid + k*1024;
  float s = 0.f;
  for (int c = 0; c < 1024; ++c) s += a[c]*W2[(size_t)c*DIM + d];
  slots[id] += s + b2[d];
}

// -------- final logits = embeddings @ slot_keys^T (WMMA) --------
__global__ __launch_bounds__(256) void logits_k(const float* __restrict__ emb,
    const float* __restrict__ sk, float* __restrict__ out) {
  const int tid = threadIdx.x;
  const int w = tid >> 5, l = tid & 31, hh = l >> 4, ln16 = l & 15;
  const size_t n0 = (size_t)blockIdx.x * 128 + w*16;
  const float* __restrict__ arow = emb + (n0 + ln16)*DIM;   // A: M=n rows
  const float* __restrict__ brow = sk + ln16*DIM;           // B: N=slot cols = sk rows
  v8f acc = zero8();
  for (int k = 0; k < DIM; k += 4) {
    v2f a  = *(const v2f*)&arow[k + hh*2];
    v2f bb = *(const v2f*)&brow[k + hh*2];
    acc = wmma4(a, bb, acc);
  }
  #pragma unroll
  for (int jj = 0; jj < 8; ++jj)
    out[(n0 + jj + hh*8)*NSLOT + ln16] = acc[jj];           // 64B coalesced per half
}

__global__ void copy_slots_k(const float* __restrict__ slots, float* __restrict__ out) {
  int i = blockIdx.x*256 + threadIdx.x;
  if (i < NSLOT*DIM) out[i] = slots[i];
}

extern "C" void kernel_launch(void* const* d_in, const int* in_sizes, int n_in,
                              void* d_out, int out_size, void* d_ws, size_t ws_size,
                              hipStream_t stream) {
  const float* emb     = (const float*)d_in[0];
  const float* noise   = (const float*)d_in[1];
  const float* mu      = (const float*)d_in[2];
  const float* sigma   = (const float*)d_in[3];
  const float* ln_in_g = (const float*)d_in[4];
  const float* ln_in_b = (const float*)d_in[5];
  const float* ln_sl_g = (const float*)d_in[6];
  const float* ln_sl_b = (const float*)d_in[7];
  const float* ln_ff_g = (const float*)d_in[8];
  const float* ln_ff_b = (const float*)d_in[9];
  const float* Wq      = (const float*)d_in[10];
  const float* Wk      = (const float*)d_in[11];
  const float* Wv      = (const float*)d_in[12];
  const float* Wih     = (const float*)d_in[13];
  const float* Whh     = (const float*)d_in[14];
  const float* bih     = (const float*)d_in[15];
  const float* bhh     = (const float*)d_in[16];
  const float* W1      = (const float*)d_in[17];
  const float* b1      = (const float*)d_in[18];
  const float* W2      = (const float*)d_in[19];
  const float* b2      = (const float*)d_in[20];
  const float* Wlg     = (const float*)d_in[21];
  (void)in_sizes; (void)n_in; (void)out_size; (void)ws_size;

  float* ws = (float*)d_ws;
  float* kproj   = ws;                                 // 67108864
  float* vproj   = kproj + (size_t)NTOK*DIM;           // 67108864
  float* attn    = vproj + (size_t)NTOK*DIM;           // 2097152
  float* slots   = attn + (size_t)NSLOT*NTOK;          // 8192
  float* qbuf    = slots + NSLOT*DIM;                  // 8192
  float* lnbuf   = qbuf + NSLOT*DIM;                   // 8192
  float* gi      = lnbuf + NSLOT*DIM;                  // 24576
  float* gh      = gi + NSLOT*3*DIM;                   // 24576
  float* upd     = gh + NSLOT*3*DIM;                   // 8192
  float* hid     = upd + NSLOT*DIM;                    // 16384
  float* skeys   = hid + NSLOT*2*DIM;                  // 8192
  float* rowsum  = skeys + NSLOT*DIM;                  // 16
  float* rspart  = rowsum + 16;                        // 16384
  float* updpart = rspart + 1024*16;                   // 1048576
  float* outf = (float*)d_out;

  init_slots_k<<<32, 256, 0, stream>>>(noise, mu, sigma, slots);
  kvproj_k<<<NTOK/16, 256, 0, stream>>>(emb, ln_in_g, ln_in_b, Wk, Wv, kproj, vproj);

  for (int it = 0; it < NUM_ITER; ++it) {
    ln_rows_k<<<16, 256, 0, stream>>>(slots, ln_sl_g, ln_sl_b, lnbuf);
    mm512_k<<<32, 256, 0, stream>>>(lnbuf, Wq, qbuf);
    attn_k<<<1024, 256, 0, stream>>>(kproj, qbuf, attn, rspart);
    reduce_rowsum_k<<<16, 256, 0, stream>>>(rspart, rowsum);
    updates_k<<<128, 256, 0, stream>>>(attn, vproj, updpart);
    reduce_upd_k<<<32, 256, 0, stream>>>(updpart, upd);
    gigh_k<<<192, 256, 0, stream>>>(upd, rowsum, slots, Wih, Whh, bih, bhh, gi, gh);
    gru_elem_k<<<32, 256, 0, stream>>>(gi, gh, slots);
    ln_rows_k<<<16, 256, 0, stream>>>(slots, ln_ff_g, ln_ff_b, lnbuf);
    ffn1_k<<<64, 256, 0, stream>>>(lnbuf, W1, b1, hid);
    ffn2_k<<<32, 256, 0, stream>>>(hid, W2, b2, slots);
  }

  mm512_k<<<32, 256, 0, stream>>>(slots, Wlg, skeys);
  logits_k<<<1024, 256, 0, stream>>>(emb, skeys, outf);
  copy_slots_k<<<32, 256, 0, stream>>>(slots, outf + (size_t)NTOK*NSLOT);
}